// LSTMHGTTagger_nofc_v1_74079595921479
// MI455X (gfx1250) — compile-verified
//
#include <hip/hip_runtime.h>

typedef float v2f __attribute__((ext_vector_type(2)));
typedef float v8f __attribute__((ext_vector_type(8)));

#define BB    64
#define TT    2048
#define EE    256
#define ROWS  (BB * TT)
#define GATES 4
#define CT    16            // timesteps per LDS chunk
#define NC    (TT / CT)     // 128 chunks

__device__ __forceinline__ float fast_sigmoid(float x) {
  float e = __expf(-x);                       // v_exp_f32
  return __builtin_amdgcn_rcpf(1.0f + e);     // v_rcp_f32
}

__device__ __forceinline__ float fast_tanh(float x) {
  x = fminf(fmaxf(x, -20.0f), 20.0f);         // keep exp finite
  float e = __expf(-2.0f * x);
  return (1.0f - e) * __builtin_amdgcn_rcpf(1.0f + e);
}

// ---------------------------------------------------------------------------
// Kernel 1: xg[row][g] = dot(data[row,:], w_ih[g,:]) + b_ih[g] + b_hh[g]
// One wave per 16-row tile; K swept 4-at-a-time with V_WMMA_F32_16X16X4_F32.
// A (16x4 f32): lane l<16 -> row l, K={k,k+1}; lane l+16 -> row l, K={k+2,k+3}
// B (4x16 f32): mirrored; lanes with N>=4 carry duplicates of gate rows
//   (l&3) -- those D columns are garbage-but-finite and are never stored,
//   so no masking is needed in the hot loop.
// D (16x16 f32): vgpr r, lane l<16 -> (M=r, N=l); lane l+16 -> (M=r+8, N=l).
// ---------------------------------------------------------------------------
__global__ void __launch_bounds__(256)
xg_gemm_wmma(const float* __restrict__ data,
             const float* __restrict__ w_ih,
             const float* __restrict__ b_ih,
             const float* __restrict__ b_hh,
             float* __restrict__ xg) {
  const int lane = threadIdx.x & 31;
  const int wave = threadIdx.x >> 5;
  const int tile = blockIdx.x * (blockDim.x >> 5) + wave;
  const int row0 = tile * 16;
  const bool hi  = lane >= 16;
  const int  l   = lane & 15;

  const v2f* aptr = (const v2f*)(data + (size_t)(row0 + l) * EE + (hi ? 2 : 0));
  const v2f* bptr = (const v2f*)(w_ih + (size_t)(l & 3) * EE + (hi ? 2 : 0));

  // Fold bias into the accumulator: every element this lane holds has N = l.
  const float bias = (l < 4) ? (b_ih[l] + b_hh[l]) : 0.0f;
  v8f acc;
#pragma unroll
  for (int r = 0; r < 8; ++r) acc[r] = bias;

#pragma unroll 8
  for (int k = 0; k < EE / 4; ++k) {
    v2f a = __builtin_nontemporal_load(aptr + 2 * k);  // streamed once: NT
    v2f b = bptr[2 * k];                               // reused: keep cached
    acc = __builtin_amdgcn_wmma_f32_16x16x4_f32(
        /*neg_a=*/false, a, /*neg_b=*/false, b,
        /*c_mod=*/(short)0, acc, /*reuse_a=*/false, /*reuse_b=*/false);
  }

  if (l < 4) {
    const int rbase = row0 + (hi ? 8 : 0);
#pragma unroll
    for (int r = 0; r < 8; ++r)
      xg[(size_t)(rbase + r) * GATES + l] = acc[r];
  }
}

// ---------------------------------------------------------------------------
// Kernel 2: serial LSTM scan. Lanes 0..63 each own one batch's recurrence;
// all 256 threads double-buffer xg chunks (global -> regs -> LDS) and write
// the masked-sigmoid output.  Single block: the chain is inherently serial.
// ---------------------------------------------------------------------------
__global__ void __launch_bounds__(256)
lstm_scan(const float* __restrict__ xg,
          const int* __restrict__ seq_length,
          const float* __restrict__ w_hh,
          float* __restrict__ out) {
  __shared__ float4 xbuf[2][CT][BB];   // 32 KB
  __shared__ float  hbuf[CT][BB];      // 4 KB

  const int tid = threadIdx.x;

  // seq_length passthrough (tuple output, bit-copied after the 131072 floats)
  if (tid < BB) {
    ((int*)out)[ROWS + tid] = seq_length[tid];
  }

  const float4* xg4 = (const float4*)xg;   // one float4 = 4 gates of one (b,t)

  const float w0 = w_hh[0], w1 = w_hh[1], w2 = w_hh[2], w3 = w_hh[3];

  int lenb[4];
#pragma unroll
  for (int r = 0; r < 4; ++r) {
    int e = r * 256 + tid;
    lenb[r] = seq_length[e >> 4];
  }

  // preload chunk 0
  float4 reg[4];
#pragma unroll
  for (int r = 0; r < 4; ++r) {
    int e = r * 256 + tid;
    int b = e >> 4, tl = e & (CT - 1);
    reg[r] = xg4[(size_t)b * TT + tl];
  }
#pragma unroll
  for (int r = 0; r < 4; ++r) {
    int e = r * 256 + tid;
    int b = e >> 4, tl = e & (CT - 1);
    xbuf[0][tl][b] = reg[r];
  }
  __syncthreads();

  float h = 0.0f, cst = 0.0f;
  const int myb = tid;

  for (int c = 0; c < NC; ++c) {
    const int t0  = c * CT;
    const int buf = c & 1;

    // issue prefetch of chunk c+1 (latency hidden behind the recurrence)
    if (c + 1 < NC) {
      const int tn = t0 + CT;
#pragma unroll
      for (int r = 0; r < 4; ++r) {
        int e = r * 256 + tid;
        int b = e >> 4, tl = e & (CT - 1);
        reg[r] = xg4[(size_t)b * TT + tn + tl];
      }
    }

    // serial recurrence, lanes 0..63 (one batch per lane, SIMD across batches)
    if (tid < BB) {
#pragma unroll
      for (int tl = 0; tl < CT; ++tl) {
        float4 x = xbuf[buf][tl][myb];
        float gi = fmaf(h, w0, x.x);
        float gf = fmaf(h, w1, x.y);
        float gc = fmaf(h, w2, x.z);
        float go = fmaf(h, w3, x.w);
        float ii = fast_sigmoid(gi);
        float ff = fast_sigmoid(gf);
        float gg = fast_tanh(gc);
        float oo = fast_sigmoid(go);
        cst = fmaf(ff, cst, ii * gg);
        h   = oo * fast_tanh(cst);
        hbuf[tl][myb] = h;
      }
    }
    __syncthreads();   // hbuf ready; xbuf[buf] fully consumed

    // drain prefetch into the other buffer
    if (c + 1 < NC) {
#pragma unroll
      for (int r = 0; r < 4; ++r) {
        int e = r * 256 + tid;
        int b = e >> 4, tl = e & (CT - 1);
        xbuf[buf ^ 1][tl][b] = reg[r];
      }
    }

    // masked-sigmoid writeout of chunk c (coalesced: consecutive t per batch)
#pragma unroll
    for (int r = 0; r < 4; ++r) {
      int e = r * 256 + tid;
      int b = e >> 4, tl = e & (CT - 1);
      int t = t0 + tl;
      float hv = hbuf[tl][b];
      out[(size_t)b * TT + t] = (t < lenb[r]) ? fast_sigmoid(hv) : 0.5f;
    }
    __syncthreads();   // xbuf[buf^1] ready, hbuf free
  }
}

// ---------------------------------------------------------------------------
extern "C" void kernel_launch(void* const* d_in, const int* in_sizes, int n_in,
                              void* d_out, int out_size, void* d_ws, size_t ws_size,
                              hipStream_t stream) {
  const float* data       = (const float*)d_in[0];   // (64, 2048, 256) f32
  const int*   seq_length = (const int*)d_in[1];     // (64,) i32
  const float* w_ih       = (const float*)d_in[2];   // (4, 256) f32
  const float* w_hh       = (const float*)d_in[3];   // (4, 1) f32
  const float* b_ih       = (const float*)d_in[4];   // (4,) f32
  const float* b_hh       = (const float*)d_in[5];   // (4,) f32
  float* out = (float*)d_out;                        // 131072 f32 + 64 i32
  float* xg  = (float*)d_ws;                         // ROWS*4 f32 = 2 MB

  const int waves_per_block = 8;                     // 256 threads
  const int tiles = ROWS / 16;                       // 8192
  dim3 blk(256);
  dim3 grd(tiles / waves_per_block);                 // 1024 blocks

  xg_gemm_wmma<<<grd, blk, 0, stream>>>(data, w_ih, b_ih, b_hh, xg);
  lstm_scan<<<1, 256, 0, stream>>>(xg, seq_length, w_hh, out);
}